// MinimalRNNCell_31765578121819
// MI455X (gfx1250) — compile-verified
//
#include <hip/hip_runtime.h>

// MinimalRNN, MI455X / gfx1250 (wave32, WMMA).
// Only batch row B-1 contributes to the output -> compute just that row.
// Phase 1 (parallel, WMMA f32 16x16x4): z = x@Win^T + b_in ; a = z@Wgz^T + b_g
// Phase 2 (1 workgroup, latency-optimized scan): h_t = u*h + (1-u)*z,
//         u = sigmoid(Wgh@h + a). Pair-split dot + shfl_xor reduction,
//         double-buffered h in LDS (ONE barrier/step), 4-deep a/z prefetch.
// Phase 3 (parallel, WMMA f32 16x16x4): y = h@Wout^T + b_out

typedef __attribute__((ext_vector_type(2))) float v2f;
typedef __attribute__((ext_vector_type(8))) float v8f;

#define Bsz 64
#define Tn  4096
#define In  64
#define Hn  128
#define On  64

__device__ __forceinline__ v8f wmma_f32_k4(v2f a, v2f b, v8f c) {
  // V_WMMA_F32_16X16X4_F32: D = A(16x4 f32) * B(4x16 f32) + C(16x16 f32)
  return __builtin_amdgcn_wmma_f32_16x16x4_f32(
      /*neg_a=*/false, a, /*neg_b=*/false, b,
      /*c_mod=*/(short)0, c, /*reuse_a=*/false, /*reuse_b=*/false);
}

// ---------------------------------------------------------------------------
// Phase 1: per block: 16 timesteps, 8 waves, wave w owns N-columns [16w,16w+16)
// GEMM1 (K=64): z_tile = x_tile @ Win^T + b_in  -> LDS + global zbuf
// GEMM2 (K=128): a_tile = z_tile @ Wgz^T + b_g  -> global abuf
// ---------------------------------------------------------------------------
__global__ void __launch_bounds__(256)
rnn_phase1(const float* __restrict__ x, const float* __restrict__ Win,
           const float* __restrict__ bin, const float* __restrict__ Wg,
           const float* __restrict__ bg, float* __restrict__ zbuf,
           float* __restrict__ abuf) {
  __shared__ float zt[16][Hn + 4];           // z tile, padded vs bank conflicts
  const int lane = threadIdx.x & 31;
  const int wave = threadIdx.x >> 5;         // 0..7
  const int n0   = wave * 16;
  const int row  = lane & 15;                // M for A, N for B/C/D
  const int hi   = lane >> 4;                // lane group 0/1
  const int ko   = hi * 2;                   // K offset within 4-chunk
  const int t0   = blockIdx.x * 16;
  const float* xrow = x + ((size_t)(Bsz - 1) * Tn + t0) * In;  // batch B-1 only

  // ---- GEMM 1: z[t,n] = sum_i x[t,i] * Win[n,i] ----
  v8f c = {};
#pragma unroll
  for (int kc = 0; kc < In / 4; ++kc) {
    const int kb = kc * 4;
    v2f a, b;
    a.x = xrow[row * In + kb + ko];          // A: lane row M, K = kb+ko, kb+ko+1
    a.y = xrow[row * In + kb + ko + 1];
    const int n = n0 + row;                  // B: lane col N
    b.x = Win[n * In + kb + ko];
    b.y = Win[n * In + kb + ko + 1];
    c = wmma_f32_k4(a, b, c);
  }
  {
    const int n = n0 + row;
    const float bias = bin[n];
#pragma unroll
    for (int r = 0; r < 8; ++r) {
      const int m = r + hi * 8;              // C/D layout: VGPR r -> M = r (+8)
      const float v = c[r] + bias;
      zt[m][n] = v;
      zbuf[(size_t)(t0 + m) * Hn + n] = v;
    }
  }
  __syncthreads();                           // z tile visible to all 8 waves

  // ---- GEMM 2: a[t,n] = sum_k z[t,k] * Wg[n, H+k] ----
  v8f c2 = {};
#pragma unroll
  for (int kc = 0; kc < Hn / 4; ++kc) {
    const int kb = kc * 4;
    v2f a, b;
    a.x = zt[row][kb + ko];                  // A from LDS z tile
    a.y = zt[row][kb + ko + 1];
    const int n = n0 + row;
    b.x = Wg[n * (2 * Hn) + Hn + kb + ko];   // Wgz half of W_g
    b.y = Wg[n * (2 * Hn) + Hn + kb + ko + 1];
    c2 = wmma_f32_k4(a, b, c2);
  }
  {
    const int n = n0 + row;
    const float bias = bg[n];
#pragma unroll
    for (int r = 0; r < 8; ++r) {
      const int m = r + hi * 8;
      abuf[(size_t)(t0 + m) * Hn + n] = c2[r] + bias;
    }
  }
}

// ---------------------------------------------------------------------------
// Phase 2: sequential scan, one workgroup (256 threads = 8 wave32).
// Thread tid = 2j+half: hidden unit j (0..127), k-half (0/1).
// Wgh row halves pinned in VGPRs (64 floats/thread = full 128x128 in the WG).
// Dot: 64 FMAs + __shfl_xor pair reduction (lanes 2j,2j+1 share a wave).
// h double-buffered in LDS -> a single barrier per step.
// a/z streamed through a 4-deep register prefetch queue (hides L2 latency).
// ---------------------------------------------------------------------------
__global__ void __launch_bounds__(256)
rnn_phase2(const float* __restrict__ Wg, const float* __restrict__ zbuf,
           const float* __restrict__ abuf, float* __restrict__ hbuf) {
  __shared__ float hsh[2][Hn];
  const int tid  = threadIdx.x;
  const int j    = tid >> 1;                 // hidden unit 0..127
  const int half = tid & 1;                  // k in [64*half, 64*half+64)

  float w[64];                               // Wgh[j][half*64 + k] in VGPRs
#pragma unroll
  for (int k = 0; k < 64; ++k)
    w[k] = Wg[(size_t)j * (2 * Hn) + half * 64 + k];

  if (tid < Hn) hsh[0][tid] = 0.0f;          // h0 = 0
  __syncthreads();

  // 4-deep prefetch queues (consumed by half==0 threads only)
  float a0 = 0.f, a1 = 0.f, a2 = 0.f, a3 = 0.f;
  float z0 = 0.f, z1 = 0.f, z2 = 0.f, z3 = 0.f;
  if (half == 0) {
    a0 = abuf[0 * Hn + j];  z0 = zbuf[0 * Hn + j];
    a1 = abuf[1 * Hn + j];  z1 = zbuf[1 * Hn + j];
    a2 = abuf[2 * Hn + j];  z2 = zbuf[2 * Hn + j];
    a3 = abuf[3 * Hn + j];  z3 = zbuf[3 * Hn + j];
  }

#define RNN_STEP(tt, ac, zc)                                                 \
  do {                                                                       \
    const int cur = (tt) & 1;                                                \
    float p = 0.0f;                                                          \
    _Pragma("unroll")                                                        \
    for (int k = 0; k < 64; ++k)                                             \
      p += w[k] * hsh[cur][half * 64 + k];                                   \
    p += __shfl_xor(p, 1, 32);               /* pair-sum within wave */      \
    if (half == 0) {                                                         \
      const float pre = (ac) + p;                                            \
      const float u   = 1.0f / (1.0f + __expf(-pre));                        \
      const float hn  = u * hsh[cur][j] + (1.0f - u) * (zc);                 \
      hsh[cur ^ 1][j] = hn;                  /* write other buffer */        \
      hbuf[(size_t)(tt) * Hn + j] = hn;                                      \
      if ((tt) + 4 < Tn) {                   /* refill queue slot */         \
        (ac) = abuf[(size_t)((tt) + 4) * Hn + j];                            \
        (zc) = zbuf[(size_t)((tt) + 4) * Hn + j];                            \
      }                                                                      \
    }                                                                        \
    __syncthreads();                         /* one barrier per step */      \
  } while (0)

  for (int t = 0; t < Tn; t += 4) {
    RNN_STEP(t + 0, a0, z0);
    RNN_STEP(t + 1, a1, z1);
    RNN_STEP(t + 2, a2, z2);
    RNN_STEP(t + 3, a3, z3);
  }
#undef RNN_STEP
}

// ---------------------------------------------------------------------------
// Phase 3: y = h @ Wout^T + b_out, [T,128]x[128,64] via WMMA f32 16x16x4.
// Per block: 16 timesteps, 4 waves, wave w owns output columns [16w,16w+16).
// ---------------------------------------------------------------------------
__global__ void __launch_bounds__(128)
rnn_phase3(const float* __restrict__ hbuf, const float* __restrict__ Wout,
           const float* __restrict__ bout, float* __restrict__ y) {
  const int lane = threadIdx.x & 31;
  const int wave = threadIdx.x >> 5;         // 0..3
  const int n0   = wave * 16;
  const int row  = lane & 15;
  const int hi   = lane >> 4;
  const int ko   = hi * 2;
  const int t0   = blockIdx.x * 16;

  v8f c = {};
#pragma unroll
  for (int kc = 0; kc < Hn / 4; ++kc) {
    const int kb = kc * 4;
    v2f a, b;
    a.x = hbuf[(size_t)(t0 + row) * Hn + kb + ko];
    a.y = hbuf[(size_t)(t0 + row) * Hn + kb + ko + 1];
    const int n = n0 + row;
    b.x = Wout[n * Hn + kb + ko];
    b.y = Wout[n * Hn + kb + ko + 1];
    c = wmma_f32_k4(a, b, c);
  }
  const int n = n0 + row;
  const float bias = bout[n];
#pragma unroll
  for (int r = 0; r < 8; ++r) {
    const int m = r + hi * 8;
    y[(size_t)(t0 + m) * On + n] = c[r] + bias;
  }
}

// ---------------------------------------------------------------------------
extern "C" void kernel_launch(void* const* d_in, const int* in_sizes, int n_in,
                              void* d_out, int out_size, void* d_ws, size_t ws_size,
                              hipStream_t stream) {
  const float* x    = (const float*)d_in[0];  // [B,T,I]
  const float* Win  = (const float*)d_in[1];  // [H,I]
  const float* bin  = (const float*)d_in[2];  // [H]
  const float* Wg   = (const float*)d_in[3];  // [H,2H]
  const float* bg   = (const float*)d_in[4];  // [H]
  const float* Wout = (const float*)d_in[5];  // [O,H]
  const float* bout = (const float*)d_in[6];  // [O]

  float* zbuf = (float*)d_ws;                 // [T,H]  2 MB
  float* abuf = zbuf + (size_t)Tn * Hn;       // [T,H]  2 MB
  float* hbuf = abuf + (size_t)Tn * Hn;       // [T,H]  2 MB
  float* yout = (float*)d_out;                // [T,O]

  rnn_phase1<<<Tn / 16, 256, 0, stream>>>(x, Win, bin, Wg, bg, zbuf, abuf);
  rnn_phase2<<<1, 256, 0, stream>>>(Wg, zbuf, abuf, hbuf);
  rnn_phase3<<<Tn / 16, 128, 0, stream>>>(hbuf, Wout, bout, yout);
}